// Normalizer_6640019440027
// MI455X (gfx1250) — compile-verified
//
#include <hip/hip_runtime.h>
#include <stdint.h>

// Row-normalize: out[row][j] = adj[row][j] * inv(sum_j adj[row][j])
// Shapes: [2, 8, 2048, 2048] fp32 -> 32768 rows of 2048 floats.
// Memory-bound: 512 MB total traffic -> ~22us at 23.3 TB/s roofline.
// One block (256 threads = 8 wave32) per row; single pass via LDS staging
// using the gfx1250 async global->LDS data mover (ASYNCcnt path).
//
// The two async copies + counter waits are fused into ONE asm block so the
// register allocator cannot reuse the address VGPRs while the async op is in
// flight (XNACK replay may re-read VMEM address operands; XCNT hazard).

#define ROW_LEN 2048
#define BLOCK   256
#define WAVES   (BLOCK / 32)

__global__ __launch_bounds__(BLOCK) void row_normalize_kernel(
    const float* __restrict__ adj, float* __restrict__ out)
{
    __shared__ __align__(16) float smem[ROW_LEN];   // 8 KB row stage
    __shared__ float wsum[WAVES];

    const int tid = threadIdx.x;
    const size_t row = blockIdx.x;

    const float* src = adj + row * (size_t)ROW_LEN;
    float*       dst = out + row * (size_t)ROW_LEN;

    // ---- Stage row into LDS with async global->LDS b128 copies (CDNA5) ----
    // Thread t owns float4 slots t and t+256 of the row (coalesced b128).
    const uint64_t g0 = (uint64_t)(uintptr_t)src + (uint64_t)tid * 16u;
    const uint64_t g1 = g0 + (uint64_t)BLOCK * 16u;
    // Low 32 bits of the flat address of a __shared__ object are the
    // workgroup-relative LDS byte offset (ISA 10.2 aperture mapping).
    const uint32_t l0 = (uint32_t)(uintptr_t)(&smem[0]) + (uint32_t)tid * 16u;
    const uint32_t l1 = l0 + (uint32_t)BLOCK * 16u;

    // Single asm block: issue both async copies, then drain XCNT (address
    // translation) and ASYNCcnt (data landed in LDS) before releasing any
    // operand registers back to the allocator.
    asm volatile(
        "global_load_async_to_lds_b128 %0, %2, off\n\t"
        "global_load_async_to_lds_b128 %1, %3, off\n\t"
        "s_wait_xcnt 0x0\n\t"
        "s_wait_asynccnt 0x0"
        :: "v"(l0), "v"(l1), "v"(g0), "v"(g1) : "memory");

    // All waves' copies complete before the cross-wave reduction below.
    __syncthreads();

    // ---- Read own 8 floats back from LDS (ds_load_b128 x2) ----
    const float4* sm4 = (const float4*)smem;
    float4 a0 = sm4[tid];
    float4 a1 = sm4[tid + BLOCK];

    // ---- Row sum: per-thread partial -> wave32 butterfly -> block total ----
    float s = (a0.x + a0.y) + (a0.z + a0.w) + (a1.x + a1.y) + (a1.z + a1.w);
    #pragma unroll
    for (int m = 16; m > 0; m >>= 1)
        s += __shfl_xor(s, m, 32);
    if ((tid & 31) == 0) wsum[tid >> 5] = s;
    __syncthreads();

    float total = 0.0f;
    #pragma unroll
    for (int w = 0; w < WAVES; ++w) total += wsum[w];

    // inv = 1/degree with NaN/Inf -> 0 (matches jnp.where(isfinite(inv),...))
    float inv = 1.0f / total;
    if (!__builtin_isfinite(inv)) inv = 0.0f;

    // ---- Scale and store (global_store_b128 x2, coalesced) ----
    float4 r0 = make_float4(a0.x * inv, a0.y * inv, a0.z * inv, a0.w * inv);
    float4 r1 = make_float4(a1.x * inv, a1.y * inv, a1.z * inv, a1.w * inv);
    float4* d4 = (float4*)dst;
    d4[tid]         = r0;
    d4[tid + BLOCK] = r1;
}

extern "C" void kernel_launch(void* const* d_in, const int* in_sizes, int n_in,
                              void* d_out, int out_size, void* d_ws, size_t ws_size,
                              hipStream_t stream)
{
    const float* adj = (const float*)d_in[0];
    float*       out = (float*)d_out;

    const int total_elems = in_sizes[0];           // 67,108,864
    const int nrows = total_elems / ROW_LEN;       // 32,768

    hipLaunchKernelGGL(row_normalize_kernel,
                       dim3(nrows), dim3(BLOCK), 0, stream,
                       adj, out);
}